// SequenceModel_71519795413523
// MI455X (gfx1250) — compile-verified
//
#include <hip/hip_runtime.h>

// CDNA5 (gfx1250) wave32 WMMA types
typedef __attribute__((ext_vector_type(2))) float v2f;   // A/B of V_WMMA_F32_16X16X4_F32
typedef __attribute__((ext_vector_type(8))) float v8f;   // C/D 16x16 f32

#define B_   1024
#define T_   2048
#define D_   10
#define H_   64
#define ROW  68   // padded LDS row stride (floats): banks = (4*M + off) % 64 -> conflict-free

// tanh via v_exp_f32 / v_rcp_f32, computed on |x| so exp never overflows:
// tanh(a) = (1 - e)/(1 + e), e = exp2(-2a*log2 e); then restore sign.
__device__ __forceinline__ float fast_tanh(float x) {
    float a = __builtin_fabsf(x);
    float e = __builtin_amdgcn_exp2f(a * -2.8853900817779268f);
    float r = (1.0f - e) * __builtin_amdgcn_rcpf(1.0f + e);
    return __builtin_copysignf(r, x);
}

__global__ __launch_bounds__(64) void rnn_scan_kernel(
    const float* __restrict__ xin,    // [B,T,D]
    const int*   __restrict__ slen,   // [B]
    const float* __restrict__ Wxh,    // [D,H]
    const float* __restrict__ Whh,    // [H,H]
    const float* __restrict__ Wlin,   // [H,1]
    const float* __restrict__ blin,   // [1]
    float* __restrict__ out)          // [B,1]
{
    __shared__ float hbuf[16 * ROW];

    const int tid   = threadIdx.x;
    const int wid   = tid >> 5;      // wave 0 -> N-tiles 0,1 ; wave 1 -> N-tiles 2,3
    const int lane  = tid & 31;
    const int hi    = lane >> 4;     // half-wave select (K split / M split)
    const int lm    = lane & 15;
    const int bBase = blockIdx.x * 16;

    // h0 = 0
    for (int i = tid; i < 16 * ROW; i += 64) hbuf[i] = 0.0f;

    // seq lengths for the D-layout rows this lane covers (VGPR j -> batch row j + 8*hi)
    int sl[8];
#pragma unroll
    for (int j = 0; j < 8; ++j) sl[j] = slen[bBase + j + hi * 8];

    // ---- Preload B-format weights (B 4x16: VGPR v, half hi -> K = v + 2*hi) ----
    v2f bhh[2][16];
    v2f bxh[2][3];
#pragma unroll
    for (int tt = 0; tt < 2; ++tt) {
        const int col = (wid * 2 + tt) * 16 + lm;
#pragma unroll
        for (int c = 0; c < 16; ++c) {
            const int k = 4 * c + 2 * hi;
            bhh[tt][c].x = Whh[(k + 0) * H_ + col];
            bhh[tt][c].y = Whh[(k + 1) * H_ + col];
        }
#pragma unroll
        for (int c = 0; c < 2; ++c) {
            const int k = 4 * c + 2 * hi;
            bxh[tt][c].x = Wxh[(k + 0) * H_ + col];
            bxh[tt][c].y = Wxh[(k + 1) * H_ + col];
        }
        // K-chunk 2 covers rows 8..11 of W_xh; rows 10,11 are padding -> 0
        bxh[tt][2].x = hi ? 0.0f : Wxh[8 * H_ + col];
        bxh[tt][2].y = hi ? 0.0f : Wxh[9 * H_ + col];
    }

    // A-format x source: lane lm = batch row M; chunks at d = {0,4}+2*hi, and d=8 (lo half only)
    const float* xrowA = xin + ((size_t)(bBase + lm) * T_) * D_ + 2 * hi;
    const float* xrowC = xin + ((size_t)(bBase + lm) * T_) * D_ + 8;  // always in-bounds

    __syncthreads();

    for (int t = 0; t < T_; ++t) {
        // ---- x A-chunks for this step (issue early; latency hidden by h WMMAs) ----
        const float* xpA = xrowA + (size_t)t * D_;
        const float* xpC = xrowC + (size_t)t * D_;
        v2f ax[3];
        ax[0] = *(const v2f*)(xpA + 0);
        ax[1] = *(const v2f*)(xpA + 4);
        v2f c2 = *(const v2f*)(xpC);          // d = 8,9
        if (hi) { c2.x = 0.0f; c2.y = 0.0f; } // upper half covers padded d = 10,11
        ax[2] = c2;
        if (t + 1 < T_) __builtin_prefetch(xpA + D_, 0, 0);  // global_prefetch next step

        // ---- h A-chunks from LDS (16 x ds_load_b64, bank-conflict-free) ----
        v2f ah[16];
#pragma unroll
        for (int c = 0; c < 16; ++c)
            ah[c] = *(const v2f*)&hbuf[lm * ROW + 4 * c + 2 * hi];

        // ---- 38 x V_WMMA_F32_16X16X4_F32: 4 independent accumulation chains ----
        v8f z8 = {};
        v8f acc[2][2];
        acc[0][0] = z8; acc[0][1] = z8; acc[1][0] = z8; acc[1][1] = z8;
#pragma unroll
        for (int c = 0; c < 16; ++c) {
            acc[0][c & 1] = __builtin_amdgcn_wmma_f32_16x16x4_f32(
                false, ah[c], false, bhh[0][c], (short)0, acc[0][c & 1], false, false);
            acc[1][c & 1] = __builtin_amdgcn_wmma_f32_16x16x4_f32(
                false, ah[c], false, bhh[1][c], (short)0, acc[1][c & 1], false, false);
        }
#pragma unroll
        for (int c = 0; c < 3; ++c) {
            acc[0][c & 1] = __builtin_amdgcn_wmma_f32_16x16x4_f32(
                false, ax[c], false, bxh[0][c], (short)0, acc[0][c & 1], false, false);
            acc[1][c & 1] = __builtin_amdgcn_wmma_f32_16x16x4_f32(
                false, ax[c], false, bxh[1][c], (short)0, acc[1][c & 1], false, false);
        }

        __syncthreads();  // every wave has consumed h_t from LDS

        // ---- tanh + masked writeback (C/D layout: VGPR j -> batch row j + 8*hi) ----
#pragma unroll
        for (int tt = 0; tt < 2; ++tt) {
            const int col = (wid * 2 + tt) * 16 + lm;
#pragma unroll
            for (int j = 0; j < 8; ++j) {
                float v  = acc[tt][0][j] + acc[tt][1][j];
                float th = fast_tanh(v);
                if (t < sl[j])                      // exec-predicated ds_store:
                    hbuf[(j + 8 * hi) * ROW + col] = th;  // inactive rows stay frozen
            }
        }

        __syncthreads();  // h_{t+1} visible to all waves
    }

    // ---- epilogue: out[b] = h[b,:] . W_lin + b_lin ----
    if (tid < 16) {
        float s = blin[0];
#pragma unroll
        for (int j = 0; j < H_; ++j)
            s += hbuf[tid * ROW + j] * Wlin[j];
        out[bBase + tid] = s;
    }
}

extern "C" void kernel_launch(void* const* d_in, const int* in_sizes, int n_in,
                              void* d_out, int out_size, void* d_ws, size_t ws_size,
                              hipStream_t stream) {
    const float* xin  = (const float*)d_in[0];  // input_seq [B,T,D]
    const int*   sl   = (const int*)  d_in[1];  // seq_lengths [B]
    const float* Wxh  = (const float*)d_in[2];  // [D,H]
    const float* Whh  = (const float*)d_in[3];  // [H,H]
    const float* Wlin = (const float*)d_in[4];  // [H,1]
    const float* blin = (const float*)d_in[5];  // [1]
    float* out = (float*)d_out;                 // [B,1]

    dim3 grid(B_ / 16);   // 64 workgroups, one 16-batch WMMA M-tile each
    dim3 block(64);       // 2 waves; each wave owns 32 hidden outputs
    rnn_scan_kernel<<<grid, block, 0, stream>>>(xin, sl, Wxh, Whh, Wlin, blin, out);
}